// GNNModel_68083821576362
// MI455X (gfx1250) — compile-verified
//
#include <hip/hip_runtime.h>
#include <hip/hip_bf16.h>

typedef __attribute__((ext_vector_type(2))) float v2f;
typedef __attribute__((ext_vector_type(8))) float v8f;

#define HID 128

// ---------------- zero ----------------
__global__ void k_zero(float* __restrict__ p, long long n) {
    long long i = (long long)blockIdx.x * blockDim.x + threadIdx.x;
    long long stride = (long long)gridDim.x * blockDim.x;
    for (; i < n; i += stride) p[i] = 0.0f;
}

// ---------------- degree on dst (self-loop +1 applied by consumers) ----------------
__global__ void k_degree(const int* __restrict__ dst, float* __restrict__ deg, int E) {
    int i = blockIdx.x * blockDim.x + threadIdx.x;
    int stride = gridDim.x * blockDim.x;
    for (; i < E; i += stride) atomicAdd(&deg[dst[i]], 1.0f);
}

// ---------------- layer-1 scalar edge aggregation ----------------
__global__ void k_sagg(const int* __restrict__ src, const int* __restrict__ dst,
                       const float* __restrict__ x, const float* __restrict__ deg,
                       float* __restrict__ sagg, int E) {
    int i = blockIdx.x * blockDim.x + threadIdx.x;
    int stride = gridDim.x * blockDim.x;
    for (; i < E; i += stride) {
        int s = src[i], d = dst[i];
        float norm = rsqrtf(deg[s] + 1.0f) * rsqrtf(deg[d] + 1.0f);
        atomicAdd(&sagg[d], x[s] * norm);
    }
}

// ---------------- h1 = relu(s[i]*W1 + b1), s = sagg + x/deg ----------------
__global__ void k_h1(const float* __restrict__ x, const float* __restrict__ deg,
                     const float* __restrict__ sagg, const float* __restrict__ W1,
                     const float* __restrict__ b1, float* __restrict__ h1, int N) {
    long long idx = (long long)blockIdx.x * blockDim.x + threadIdx.x;
    long long total = (long long)N * HID;
    long long stride = (long long)gridDim.x * blockDim.x;
    for (; idx < total; idx += stride) {
        int i = (int)(idx >> 7);
        int j = (int)(idx & (HID - 1));
        float dv = deg[i] + 1.0f;
        float s = sagg[i] + x[i] / dv;
        float v = s * W1[j] + b1[j];
        h1[idx] = fmaxf(v, 0.0f);
    }
}

// ---------------- t = h1 @ W2 via V_WMMA_F32_16X16X4_F32 ----------------
// block = 256 threads = 8 waves; blockIdx.x = M-tile (16 rows); wave = N-tile (16 cols)
__global__ void k_gemm_wmma(const float* __restrict__ A, const float* __restrict__ B,
                            float* __restrict__ C, int M) {
    int wave = threadIdx.x >> 5;
    int lane = threadIdx.x & 31;
    int lr   = lane & 15;        // lane row within half
    int hi   = lane >> 4;        // 0: lanes 0-15, 1: lanes 16-31
    int m0   = blockIdx.x * 16;
    int n0   = wave * 16;

    int mrow = m0 + lr;
    if (mrow >= M) mrow = M - 1;          // clamp loads, keep EXEC all-1 for WMMA
    const float* Arow = A + (long long)mrow * HID;

    v8f acc = {};
    #pragma unroll
    for (int k = 0; k < HID; k += 4) {
        int kk = k + 2 * hi;              // lanes 0-15: K=k,k+1 ; lanes 16-31: K=k+2,k+3
        v2f a, b;
        a.x = Arow[kk];
        a.y = Arow[kk + 1];
        b.x = B[(long long)kk * HID + n0 + lr];
        b.y = B[(long long)(kk + 1) * HID + n0 + lr];
        acc = __builtin_amdgcn_wmma_f32_16x16x4_f32(
            /*neg_a=*/false, a, /*neg_b=*/false, b,
            /*c_mod=*/(short)0, acc, /*reuse_a=*/false, /*reuse_b=*/false);
    }

    // D layout: VGPR r -> row m0 + r + 8*hi, col n0 + lr
    #pragma unroll
    for (int r = 0; r < 8; ++r) {
        int m = m0 + r + 8 * hi;
        if (m < M) C[(long long)m * HID + n0 + lr] = acc[r];
    }
}

// ---------------- layer-2 edge aggregation: agg2[dst] += t[src]*norm ----------------
// one wave per edge, 4 features per lane
__global__ void k_eagg(const int* __restrict__ src, const int* __restrict__ dst,
                       const float* __restrict__ t, const float* __restrict__ deg,
                       float* __restrict__ agg2, long long E) {
    long long gid = (long long)blockIdx.x * blockDim.x + threadIdx.x;
    long long nthreads = (long long)gridDim.x * blockDim.x;
    int lane = (int)(gid & 31);
    for (long long e = gid >> 5; e < E; e += (nthreads >> 5)) {
        int s = src[e], d = dst[e];
        float norm = rsqrtf(deg[s] + 1.0f) * rsqrtf(deg[d] + 1.0f);
        const float4 tv = *(const float4*)(t + (long long)s * HID + lane * 4);
        float* out = agg2 + (long long)d * HID + lane * 4;
        atomicAdd(out + 0, tv.x * norm);
        atomicAdd(out + 1, tv.y * norm);
        atomicAdd(out + 2, tv.z * norm);
        atomicAdd(out + 3, tv.w * norm);
    }
}

// ---------------- h2 = relu(agg2 + t/deg + b2); pool by graph ----------------
__global__ void k_h2pool(const float* __restrict__ agg2, const float* __restrict__ t,
                         const float* __restrict__ deg, const float* __restrict__ b2,
                         const int* __restrict__ batch, float* __restrict__ pool,
                         float* __restrict__ cnt, int N) {
    int i = blockIdx.x;
    int j = threadIdx.x;
    if (i >= N) return;
    float dv = deg[i] + 1.0f;
    long long idx = (long long)i * HID + j;
    float v = agg2[idx] + t[idx] / dv + b2[j];
    v = fmaxf(v, 0.0f);
    int g = batch[i];
    atomicAdd(&pool[(long long)g * HID + j], v);
    if (j == 0) atomicAdd(&cnt[g], 1.0f);
}

// ---------------- mean + MLP head: one wave per graph ----------------
__global__ void k_mlp(const float* __restrict__ pool, const float* __restrict__ cnt,
                      const float* __restrict__ Wf1, const float* __restrict__ bf1,
                      const float* __restrict__ Wf2, const float* __restrict__ bf2,
                      float* __restrict__ out, int G) {
    int g = blockIdx.x;
    int l = threadIdx.x;        // 0..31
    if (g >= G) return;
    float c = fmaxf(cnt[g], 1.0f);
    float inv = 1.0f / c;
    float z = bf1[l];
    const float* pg = pool + (long long)g * HID;
    #pragma unroll 8
    for (int j = 0; j < HID; ++j)
        z += (pg[j] * inv) * Wf1[j * 32 + l];
    z = fmaxf(z, 0.0f) * Wf2[l];
    #pragma unroll
    for (int off = 16; off > 0; off >>= 1)
        z += __shfl_down(z, off, 32);
    if (l == 0) out[g] = z + bf2[0];
}

extern "C" void kernel_launch(void* const* d_in, const int* in_sizes, int n_in,
                              void* d_out, int out_size, void* d_ws, size_t ws_size,
                              hipStream_t stream) {
    const float* x   = (const float*)d_in[0];
    const int*   ei  = (const int*)d_in[1];
    const int*   bat = (const int*)d_in[2];
    const float* W1  = (const float*)d_in[3];
    const float* b1  = (const float*)d_in[4];
    const float* W2  = (const float*)d_in[5];
    const float* b2  = (const float*)d_in[6];
    const float* Wf1 = (const float*)d_in[7];
    const float* bf1 = (const float*)d_in[8];
    const float* Wf2 = (const float*)d_in[9];
    const float* bf2 = (const float*)d_in[10];
    float* out = (float*)d_out;

    const int N = in_sizes[0];           // 100000
    const int E = in_sizes[1] / 2;       // 1600000
    const int G = out_size;              // 1024
    const int* src = ei;
    const int* dst = ei + E;

    // workspace layout (floats)
    float* ws   = (float*)d_ws;
    float* deg  = ws;                                    // N
    float* sagg = deg + N;                               // N
    float* cnt  = sagg + N;                              // G
    float* pool = cnt + G;                               // G*HID
    float* bufA = pool + (long long)G * HID;             // N*HID  (h1, later agg2)
    float* bufB = bufA + (long long)N * HID;             // N*HID  (t)

    const long long smallZero = 2LL * N + G + (long long)G * HID;
    const long long bigCnt    = (long long)N * HID;

    // 1. zero deg|sagg|cnt|pool
    k_zero<<<4096, 256, 0, stream>>>(deg, smallZero);
    // 2. degree
    k_degree<<<4096, 256, 0, stream>>>(dst, deg, E);
    // 3. layer-1 scalar aggregation
    k_sagg<<<4096, 256, 0, stream>>>(src, dst, x, deg, sagg, E);
    // 4. h1
    k_h1<<<8192, 256, 0, stream>>>(x, deg, sagg, W1, b1, bufA, N);
    // 5. t = h1 @ W2 (WMMA f32 16x16x4)
    k_gemm_wmma<<<(N + 15) / 16, 256, 0, stream>>>(bufA, W2, bufB, N);
    // 6. zero agg2 (reuses bufA)
    k_zero<<<8192, 256, 0, stream>>>(bufA, bigCnt);
    // 7. layer-2 edge aggregation (wave per edge)
    {
        long long threads = (long long)E * 32;
        int blocks = (int)((threads + 255) / 256);
        if (blocks > 200000) blocks = 200000;
        k_eagg<<<blocks, 256, 0, stream>>>(src, dst, bufB, deg, bufA, E);
    }
    // 8. h2 + pooling
    k_h2pool<<<N, HID, 0, stream>>>(bufA, bufB, deg, b2, bat, pool, cnt, N);
    // 9. mean + MLP head
    k_mlp<<<G, 32, 0, stream>>>(pool, cnt, Wf1, bf1, Wf2, bf2, out, G);
}